// LogEigLayer_2284922602274
// MI455X (gfx1250) — compile-verified
//
#include <hip/hip_runtime.h>
#include <math.h>

// CDNA5 (gfx1250) batched SPD matrix-log via inverse scaling & squaring +
// Newton-Schulz square roots. All heavy math is V_WMMA_F32_16X16X4_F32.
//
// LDS layout ("K-pair interleaved"): element (r,c) lives at
//   (r>>1)*PSTRIDE + 2*c + (r&1)
// so the WMMA B-fragment pair {B[k][n], B[k+1][n]} (k even) is one aligned
// ds_load_b64 straight into an even VGPR pair -> no repacking movs.
// PSTRIDE = 288 dwords (== 32 mod 64 banks) makes the two half-waves of a
// B-fragment load use disjoint bank halves: conflict-free.

typedef __attribute__((ext_vector_type(2))) float v2f;
typedef __attribute__((ext_vector_type(4))) float v4f;
typedef __attribute__((ext_vector_type(8))) float v8f;

#define MAT_N     128
#define PSTRIDE   288                 // dwords per row-pair (256 data + 32 pad)
#define MAT_ELEMS (64 * PSTRIDE)      // 64 row-pairs
#define NTHREADS  256                 // 8 waves of 32
#define KSQRT     4                   // number of matrix square roots
#define LOG_TERMS 12                  // Taylor terms for log(I+X)

__device__ __forceinline__ int laddr(int r, int c) {
    return (r >> 1) * PSTRIDE + c * 2 + (r & 1);
}

// dst = scale*(A @ B) + addDiag*I   (LDS pointers, dst distinct from A/B).
// Each of the 8 waves computes a 32x64 block: 2 tile-rows x 4 tile-cols.
__device__ __forceinline__ void mm128(float* dst, const float* A, const float* B,
                                      float scale, float addDiag, int lane, int wave)
{
    const int tm2 = wave >> 1;        // which 32-row block   (0..3)
    const int tn2 = wave & 1;         // which 64-col block   (0..1)
    const int h   = lane >> 4;        // half-wave: 0 -> K{0,1}, 1 -> K{2,3}
    const int l   = lane & 15;

    const v8f vzero = {0.f, 0.f, 0.f, 0.f, 0.f, 0.f, 0.f, 0.f};
    v8f acc[2][4];
#pragma unroll
    for (int i = 0; i < 2; ++i)
#pragma unroll
        for (int j = 0; j < 4; ++j) acc[i][j] = vzero;

    // A fragment bases: A[rowA][k+2h] at Ab[i] + 2k ; A[rowA][k+2h+1] at +2 more
    const float* Ab[2];
#pragma unroll
    for (int i = 0; i < 2; ++i) {
        const int rowA = (tm2 * 2 + i) * 16 + l;
        Ab[i] = A + (rowA >> 1) * PSTRIDE + (rowA & 1) + 4 * h;
    }
    // B fragment bases: {B[k+2h][colB], B[k+2h+1][colB]} contiguous at Bb[j] + (k>>1)*PSTRIDE
    const float* Bb[4];
#pragma unroll
    for (int j = 0; j < 4; ++j) {
        const int colB = (tn2 * 4 + j) * 16 + l;
        Bb[j] = B + colB * 2 + h * PSTRIDE;
    }

    for (int s = 0; s < 32; ++s) {        // k = 4*s
        v2f a[2];
#pragma unroll
        for (int i = 0; i < 2; ++i) {
            a[i].x = Ab[i][8 * s];
            a[i].y = Ab[i][8 * s + 2];
        }
#pragma unroll
        for (int j = 0; j < 4; ++j) {
            const v2f b = *(const v2f*)(Bb[j] + 2 * s * PSTRIDE);  // ds_load_b64
#pragma unroll
            for (int i = 0; i < 2; ++i) {
                acc[i][j] = __builtin_amdgcn_wmma_f32_16x16x4_f32(
                    /*neg_a=*/false, a[i], /*neg_b=*/false, b,
                    /*c_mod=*/(short)0, acc[i][j],
                    /*reuse_a=*/false, /*reuse_b=*/false);
            }
        }
    }

    // C/D layout: VGPR r holds row r (lanes 0-15) and row r+8 (lanes 16-31), N = lane%16
#pragma unroll
    for (int i = 0; i < 2; ++i)
#pragma unroll
        for (int j = 0; j < 4; ++j)
#pragma unroll
            for (int r = 0; r < 8; ++r) {
                const int row = (tm2 * 2 + i) * 16 + h * 8 + r;
                const int col = (tn2 * 4 + j) * 16 + l;
                float v = acc[i][j][r] * scale;
                if (row == col) v += addDiag;
                dst[laddr(row, col)] = v;
            }
    __syncthreads();   // dst complete before next op reads it
}

__global__ __launch_bounds__(NTHREADS)
void logeig_wmma_kernel(const float* __restrict__ in, float* __restrict__ out)
{
    extern __shared__ float smem[];
    float* b0  = smem;
    float* b1  = smem + 1 * MAT_ELEMS;
    float* b2  = smem + 2 * MAT_ELEMS;
    float* b3  = smem + 3 * MAT_ELEMS;
    float* red = smem + 4 * MAT_ELEMS;   // reduction scratch

    const int tid  = threadIdx.x;
    const int lane = tid & 31;
    const int wave = tid >> 5;
    const size_t base = (size_t)blockIdx.x * (MAT_N * MAT_N);
    const float* M = in + base;
    float*       O = out + base;

    if (tid == 0) red[0] = 0.f;

    // Load M into b0 (interleaved layout), accumulate Frobenius norm^2
    float ss = 0.f;
    for (int i = tid; i < (MAT_N * MAT_N) / 4; i += NTHREADS) {
        v4f v = ((const v4f*)M)[i];               // global_load_b128
        const int e = i * 4;
        const int r = e >> 7, c = e & 127;
        const int o = laddr(r, c);                // c multiple of 4
        b0[o]     = v.x;
        b0[o + 2] = v.y;
        b0[o + 4] = v.z;
        b0[o + 6] = v.w;
        ss += v.x * v.x + v.y * v.y + v.z * v.z + v.w * v.w;
    }
    __syncthreads();
    atomicAdd(red, ss);                           // ds_add_f32
    __syncthreads();
    const float fro = sqrtf(red[0]);
    const float inv = 1.0f / fro;

    // Y = M / ||M||_F  (eigenvalues in (0,1]) ; Z = I
    for (int e = tid; e < MAT_N * MAT_N; e += NTHREADS) {
        const int r = e >> 7, c = e & 127;
        b0[laddr(r, c)] *= inv;
        b1[laddr(r, c)] = (r == c) ? 1.f : 0.f;
    }
    __syncthreads();

    float *Y = b0, *Z = b1, *T = b2, *W = b3;

    // k repeated square roots via coupled Newton-Schulz:
    //   T = 1.5 I - 0.5 Z Y ; Y <- Y T ; Z <- T Z
    const int ns_iters[KSQRT] = {18, 12, 9, 7};
    for (int r = 0; r < KSQRT; ++r) {
        if (r > 0) {                   // reset Z = I for the next sqrt
            for (int e = tid; e < MAT_N * MAT_N; e += NTHREADS) {
                const int rr = e >> 7, cc = e & 127;
                Z[laddr(rr, cc)] = (rr == cc) ? 1.f : 0.f;
            }
            __syncthreads();
        }
        for (int it = 0; it < ns_iters[r]; ++it) {
            mm128(T, Z, Y, -0.5f, 1.5f, lane, wave);  // T = 1.5I - 0.5 Z@Y
            mm128(W, Y, T,  1.0f, 0.0f, lane, wave);  // newY -> W
            mm128(Y, T, Z,  1.0f, 0.0f, lane, wave);  // newZ -> old Y buffer
            float *y = Y, *z = Z, *w = W;
            Y = w; Z = y; W = z;                       // rotate buffers
        }
    }
    // S = (M/f)^(1/2^k) now in Y.

    // X = S - I -> Z ;  R = c_m I -> T
    const float cLast = ((LOG_TERMS & 1) ? 1.f : -1.f) / (float)LOG_TERMS;
    for (int e = tid; e < MAT_N * MAT_N; e += NTHREADS) {
        const int r = e >> 7, c = e & 127;
        const float d = (r == c) ? 1.f : 0.f;
        const int o = laddr(r, c);
        Z[o] = Y[o] - d;
        T[o] = d * cLast;
    }
    __syncthreads();

    // Horner: R_j = c_j I + X @ R_{j+1},  c_j = (-1)^{j+1}/j
    for (int j = LOG_TERMS - 1; j >= 1; --j) {
        const float cj = ((j & 1) ? 1.f : -1.f) / (float)j;
        mm128(W, Z, T, 1.0f, cj, lane, wave);
        float* t = T; T = W; W = t;
    }

    // log(M) = 2^k * (X @ R_1) + log(f) * I   -> W
    mm128(W, Z, T, (float)(1 << KSQRT), logf(fro), lane, wave);

    // Store result (gather from interleaved layout, b128 to global)
    for (int i = tid; i < (MAT_N * MAT_N) / 4; i += NTHREADS) {
        const int e = i * 4;
        const int r = e >> 7, c = e & 127;
        const int o = laddr(r, c);
        v4f v;
        v.x = W[o];
        v.y = W[o + 2];
        v.z = W[o + 4];
        v.w = W[o + 6];
        ((v4f*)O)[i] = v;
    }
}

extern "C" void kernel_launch(void* const* d_in, const int* in_sizes, int n_in,
                              void* d_out, int out_size, void* d_ws, size_t ws_size,
                              hipStream_t stream)
{
    const float* M = (const float*)d_in[0];
    float* out = (float*)d_out;
    const int batch = in_sizes[0] / (MAT_N * MAT_N);   // 4096

    const size_t shmem = (size_t)(4 * MAT_ELEMS + 16) * sizeof(float); // 288 KB (CDNA5 WGP: 320 KB)
    (void)hipFuncSetAttribute((const void*)logeig_wmma_kernel,
                              hipFuncAttributeMaxDynamicSharedMemorySize, (int)shmem);

    logeig_wmma_kernel<<<batch, NTHREADS, shmem, stream>>>(M, out);
}